// DGCNN_semseg_80814104641851
// MI455X (gfx1250) — compile-verified
//
#include <hip/hip_runtime.h>
#include <cstdint>
#include <cstddef>

#define B_    8
#define C_    9
#define N_    2048
#define KNN_  20
#define BN_   (B_ * N_)        // 16384
#define BNK_  (BN_ * KNN_)     // 327680
#define EPS_  1e-5f
#define NEG_INF_ (-3.402823e38f)

typedef __attribute__((ext_vector_type(2))) float v2f;
typedef __attribute__((ext_vector_type(8))) float v8f;

__device__ __forceinline__ float lrelu_(float x) { return x >= 0.f ? x : 0.2f * x; }

// ---------------------------------------------------------------------------
// x(B,C,N) -> xt(B,N,C); also writes the first output (inputs) to d_out.
// ---------------------------------------------------------------------------
__global__ void k_transpose(const float* __restrict__ x, float* __restrict__ xt,
                            float* __restrict__ out) {
  int tid = blockIdx.x * blockDim.x + threadIdx.x;
  if (tid >= B_ * N_ * C_) return;
  int c = tid % C_;
  int n = (tid / C_) % N_;
  int b = tid / (C_ * N_);
  float v = x[((size_t)b * C_ + c) * N_ + n];
  xt[tid] = v;
  out[tid] = v;
}

// ---------------------------------------------------------------------------
// KNN: top-20 by pd = 2*dot - |q|^2 - |p|^2 (descending, stable ties).
// Block = 256 queries of one batch; candidates staged in LDS tiles of 128.
// ---------------------------------------------------------------------------
template <int DIMS>
__global__ void k_knn(const float* __restrict__ pts, int stride, int* __restrict__ idxOut) {
  const int TILE = 128;
  __shared__ float sp[TILE * DIMS];
  __shared__ float spsq[TILE];
  int b = blockIdx.x / (N_ / 256);
  int n = (blockIdx.x % (N_ / 256)) * 256 + threadIdx.x;
  const float* base = pts + (size_t)b * N_ * stride;

  float q[DIMS];
  float qsq = 0.f;
#pragma unroll
  for (int c = 0; c < DIMS; c++) {
    float t = base[(size_t)n * stride + c];
    q[c] = t;
    qsq += t * t;
  }

  float val[KNN_];
  int vid[KNN_];
#pragma unroll
  for (int j = 0; j < KNN_; j++) { val[j] = NEG_INF_; vid[j] = 0; }

  for (int t0 = 0; t0 < N_; t0 += TILE) {
    __syncthreads();
    for (int i = threadIdx.x; i < TILE * DIMS; i += 256) {
      int r = i / DIMS, c = i % DIMS;
      sp[i] = base[(size_t)(t0 + r) * stride + c];
    }
    __syncthreads();
    if (threadIdx.x < TILE) {
      float s = 0.f;
#pragma unroll
      for (int c = 0; c < DIMS; c++) { float p = sp[threadIdx.x * DIMS + c]; s += p * p; }
      spsq[threadIdx.x] = s;
    }
    __syncthreads();
    for (int r = 0; r < TILE; r++) {
      float d = 0.f;
#pragma unroll
      for (int c = 0; c < DIMS; c++) d += q[c] * sp[r * DIMS + c];
      d = 2.f * d - qsq - spsq[r];
      int m = t0 + r;
      if (d > val[KNN_ - 1]) {
        // Branch-free unrolled insertion keeps arrays in registers.
#pragma unroll
        for (int j = KNN_ - 1; j >= 1; j--) {
          float vj = val[j], vjm = val[j - 1];
          int ijm = vid[j - 1];
          if (d > vjm)      { val[j] = vjm; vid[j] = ijm; }
          else if (d > vj)  { val[j] = d;   vid[j] = m;   }
        }
        if (d > val[0]) { val[0] = d; vid[0] = m; }
      }
    }
  }
  int* o = idxOut + ((size_t)b * N_ + n) * KNN_;
#pragma unroll
  for (int j = 0; j < KNN_; j++) o[j] = vid[j];
}

// ---------------------------------------------------------------------------
// graph_feature: feat[b,n,k,:] = [src[b,idx]-src[b,n] , src[b,n]]  (2*C ch)
// ---------------------------------------------------------------------------
template <int C>
__global__ void k_edge_feat(const float* __restrict__ src, const int* __restrict__ idx,
                            float* __restrict__ feat) {
  size_t tid = (size_t)blockIdx.x * blockDim.x + threadIdx.x;
  const size_t total = (size_t)BNK_ * (2 * C);
  if (tid >= total) return;
  int c = (int)(tid % (2 * C));
  size_t bnk = tid / (2 * C);
  size_t bn = bnk / KNN_;
  size_t b = bn / N_;
  float center = src[bn * C + (c % C)];
  float v;
  if (c < C) {
    int j = idx[bnk];
    v = src[(b * N_ + (size_t)j) * C + c] - center;
  } else {
    v = center;
  }
  feat[tid] = v;
}

// ---------------------------------------------------------------------------
// Out[M,O] = A[M,Ci] * W[O,Ci]^T via V_WMMA_F32_16X16X4_F32.
// Wave computes a 32x32 tile (2 M-tiles x 2 O-tiles, 4 accumulators) so each
// A/W b64 load feeds two WMMAs. 8 waves/block stack along M sharing the W
// slice (L0/L2 reuse). Fragment layout (ISA 7.12.2): lanes 0-15 hold K{0,1},
// lanes 16-31 hold K{2,3}; C/D register r -> row (r + 8*half), col = lane&15.
// Main loop is unguarded (no divergent exec masking); a single guarded tail
// handles Ci % 4 != 0 (Ci=18). Requires M % 256 == 0, O % 32 == 0, Ci even.
// All row offsets fit in 32 bits -> pure 32-bit address math.
// ---------------------------------------------------------------------------
__global__ void k_gemm_wmma(const float* __restrict__ A, const float* __restrict__ W,
                            float* __restrict__ Out, int M, int Ci, int O) {
  int lane = threadIdx.x & 31;
  int wave = threadIdx.x >> 5;      // wave32
  int half = lane >> 4;
  int lm   = lane & 15;
  int o0 = blockIdx.x * 32;
  int m0 = blockIdx.y * 256 + wave * 32;

  const float* a0 = A + (m0 + lm) * Ci;   // rows m0+lm .. (tile 0)
  const float* a1 = a0 + 16 * Ci;         // rows m0+16+lm (tile 1)
  const float* w0 = W + (o0 + lm) * Ci;   // cols o0+lm
  const float* w1 = w0 + 16 * Ci;         // cols o0+16+lm

  v8f acc00 = {}, acc01 = {}, acc10 = {}, acc11 = {};
  const int kk0 = 2 * half;               // lane's K sub-offset within a 4-step
  const int kmain = Ci & ~3;

#pragma unroll 2
  for (int k = 0; k < kmain; k += 4) {
    v2f aA = *reinterpret_cast<const v2f*>(a0 + k + kk0);
    v2f aB = *reinterpret_cast<const v2f*>(a1 + k + kk0);
    v2f wA = *reinterpret_cast<const v2f*>(w0 + k + kk0);
    v2f wB = *reinterpret_cast<const v2f*>(w1 + k + kk0);
    acc00 = __builtin_amdgcn_wmma_f32_16x16x4_f32(false, aA, false, wA, (short)0, acc00, false, false);
    acc01 = __builtin_amdgcn_wmma_f32_16x16x4_f32(false, aA, false, wB, (short)0, acc01, false, false);
    acc10 = __builtin_amdgcn_wmma_f32_16x16x4_f32(false, aB, false, wA, (short)0, acc10, false, false);
    acc11 = __builtin_amdgcn_wmma_f32_16x16x4_f32(false, aB, false, wB, (short)0, acc11, false, false);
  }
  if (kmain < Ci) {                       // tail (Ci even: kk,kk+1 both valid)
    int kk = kmain + kk0;
    v2f aA = {0.f, 0.f}, aB = {0.f, 0.f}, wA = {0.f, 0.f}, wB = {0.f, 0.f};
    if (kk < Ci) {
      aA = *reinterpret_cast<const v2f*>(a0 + kk);
      aB = *reinterpret_cast<const v2f*>(a1 + kk);
      wA = *reinterpret_cast<const v2f*>(w0 + kk);
      wB = *reinterpret_cast<const v2f*>(w1 + kk);
    }
    acc00 = __builtin_amdgcn_wmma_f32_16x16x4_f32(false, aA, false, wA, (short)0, acc00, false, false);
    acc01 = __builtin_amdgcn_wmma_f32_16x16x4_f32(false, aA, false, wB, (short)0, acc01, false, false);
    acc10 = __builtin_amdgcn_wmma_f32_16x16x4_f32(false, aB, false, wA, (short)0, acc10, false, false);
    acc11 = __builtin_amdgcn_wmma_f32_16x16x4_f32(false, aB, false, wB, (short)0, acc11, false, false);
  }

  float* p0 = Out + m0 * O + o0 + lm;     // rows m0.., cols o0+lm
  float* p1 = p0 + 16 * O;                // rows m0+16..
#pragma unroll
  for (int r = 0; r < 8; r++) {
    int ro = (r + 8 * half) * O;
    p0[ro]      = acc00[r];
    p0[ro + 16] = acc01[r];
    p1[ro]      = acc10[r];
    p1[ro + 16] = acc11[r];
  }
}

// ---------------------------------------------------------------------------
// Batch-norm statistics: one block per channel, reduce sum/sumsq over M rows.
// ---------------------------------------------------------------------------
__global__ void k_bn_stats(const float* __restrict__ H, int M, int O,
                           float* __restrict__ mean, float* __restrict__ rstd) {
  int c = blockIdx.x;
  __shared__ float ss[256], sq[256];
  float s = 0.f, s2 = 0.f;
  for (int m = threadIdx.x; m < M; m += 256) {
    float v = H[(size_t)m * O + c];
    s += v;
    s2 += v * v;
  }
  ss[threadIdx.x] = s;
  sq[threadIdx.x] = s2;
  __syncthreads();
  for (int off = 128; off > 0; off >>= 1) {
    if (threadIdx.x < off) {
      ss[threadIdx.x] += ss[threadIdx.x + off];
      sq[threadIdx.x] += sq[threadIdx.x + off];
    }
    __syncthreads();
  }
  if (threadIdx.x == 0) {
    float mu = ss[0] / (float)M;
    float var = sq[0] / (float)M - mu * mu;
    mean[c] = mu;
    rstd[c] = rsqrtf(var + EPS_);
  }
}

// ---------------------------------------------------------------------------
// In-place BN + leaky ReLU.
// ---------------------------------------------------------------------------
__global__ void k_bn_lrelu(float* __restrict__ H, int O, size_t total,
                           const float* __restrict__ mean, const float* __restrict__ rstd,
                           const float* __restrict__ g, const float* __restrict__ bta) {
  size_t tid = (size_t)blockIdx.x * blockDim.x + threadIdx.x;
  if (tid >= total) return;
  int c = (int)(tid % O);
  float v = (H[tid] - mean[c]) * rstd[c] * g[c] + bta[c];
  H[tid] = lrelu_(v);
}

// ---------------------------------------------------------------------------
// BN + leaky ReLU + max over K neighbors: H[BNK,O] -> X[BN,O].
// ---------------------------------------------------------------------------
__global__ void k_bn_lrelu_maxk(const float* __restrict__ H, int O,
                                const float* __restrict__ mean, const float* __restrict__ rstd,
                                const float* __restrict__ g, const float* __restrict__ bta,
                                float* __restrict__ Xout) {
  size_t tid = (size_t)blockIdx.x * blockDim.x + threadIdx.x;
  const size_t total = (size_t)BN_ * O;
  if (tid >= total) return;
  int c = (int)(tid % O);
  size_t bn = tid / O;
  float mu = mean[c], rs = rstd[c] * g[c], bb = bta[c];
  float mx = NEG_INF_;
  for (int k = 0; k < KNN_; k++) {
    float v = (H[(bn * KNN_ + k) * (size_t)O + c] - mu) * rs + bb;
    mx = fmaxf(mx, lrelu_(v));
  }
  Xout[tid] = mx;
}

// ---------------------------------------------------------------------------
// Global max over N: H[B,N,1024] -> G[B,1024].
// ---------------------------------------------------------------------------
__global__ void k_gmax(const float* __restrict__ H, float* __restrict__ G) {
  int tid = blockIdx.x * blockDim.x + threadIdx.x;
  if (tid >= B_ * 1024) return;
  int c = tid % 1024;
  int b = tid / 1024;
  float mx = NEG_INF_;
  for (int n = 0; n < N_; n++) mx = fmaxf(mx, H[((size_t)b * N_ + n) * 1024 + c]);
  G[tid] = mx;
}

// ---------------------------------------------------------------------------
// Cat [x1|x2|x3] -> (BN,192)
// ---------------------------------------------------------------------------
__global__ void k_cat3(const float* __restrict__ X1, const float* __restrict__ X2,
                       const float* __restrict__ X3, float* __restrict__ Cat) {
  size_t tid = (size_t)blockIdx.x * blockDim.x + threadIdx.x;
  const size_t total = (size_t)BN_ * 192;
  if (tid >= total) return;
  int c = (int)(tid % 192);
  size_t bn = tid / 192;
  float v = (c < 64) ? X1[bn * 64 + c]
          : (c < 128) ? X2[bn * 64 + (c - 64)]
                      : X3[bn * 64 + (c - 128)];
  Cat[tid] = v;
}

// ---------------------------------------------------------------------------
// Cat [gmax(broadcast,1024)|x1|x2|x3] -> (BN,1216)
// ---------------------------------------------------------------------------
__global__ void k_cat7(const float* __restrict__ G, const float* __restrict__ X1,
                       const float* __restrict__ X2, const float* __restrict__ X3,
                       float* __restrict__ Cat) {
  size_t tid = (size_t)blockIdx.x * blockDim.x + threadIdx.x;
  const size_t total = (size_t)BN_ * 1216;
  if (tid >= total) return;
  int c = (int)(tid % 1216);
  size_t bn = tid / 1216;
  size_t b = bn / N_;
  float v;
  if (c < 1024)       v = G[b * 1024 + c];
  else if (c < 1088)  v = X1[bn * 64 + (c - 1024)];
  else if (c < 1152)  v = X2[bn * 64 + (c - 1088)];
  else                v = X3[bn * 64 + (c - 1152)];
  Cat[tid] = v;
}

// ---------------------------------------------------------------------------
// Final BN + leaky ReLU + transpose (B,N,256) -> (B,256,N) into d_out.
// ---------------------------------------------------------------------------
__global__ void k_bn_lrelu_tr_out(const float* __restrict__ H,
                                  const float* __restrict__ mean, const float* __restrict__ rstd,
                                  const float* __restrict__ g, const float* __restrict__ bta,
                                  float* __restrict__ out) {
  size_t tid = (size_t)blockIdx.x * blockDim.x + threadIdx.x;
  const size_t total = (size_t)BN_ * 256;
  if (tid >= total) return;
  int c = (int)(tid % 256);
  size_t bn = tid / 256;
  size_t b = bn / N_;
  size_t n = bn % N_;
  float v = lrelu_((H[tid] - mean[c]) * rstd[c] * g[c] + bta[c]);
  out[(b * 256 + (size_t)c) * N_ + n] = v;
}

// ---------------------------------------------------------------------------
extern "C" void kernel_launch(void* const* d_in, const int* in_sizes, int n_in,
                              void* d_out, int out_size, void* d_ws, size_t ws_size,
                              hipStream_t stream) {
  (void)in_sizes; (void)n_in; (void)out_size; (void)ws_size;
  const float* x = (const float*)d_in[0];
  const float* W[8]; const float* G[8]; const float* Bt[8];
  for (int i = 0; i < 8; i++) {
    W[i]  = (const float*)d_in[1 + 3 * i];
    G[i]  = (const float*)d_in[2 + 3 * i];
    Bt[i] = (const float*)d_in[3 + 3 * i];
  }
  float* out = (float*)d_out;
  float* fw  = (float*)d_ws;

  // Workspace layout (float offsets), ~279 MB total, BUFA/BUFB ping-pong.
  float* XT   = fw;                  // 147456
  int*   IDX  = (int*)(fw + 147456); // 327680 ints (reused for all 3 KNNs)
  float* X1   = fw + 475136;         // 1048576
  float* X2   = fw + 1523712;        // 1048576
  float* X3   = fw + 2572288;        // 1048576
  float* CAT6 = fw + 3620864;        // 3145728
  float* MEAN = fw + 6766592;        // 1024
  float* RSTD = fw + 6767616;        // 1024
  float* GMAX = fw + 6768640;        // 8192
  float* BUFA = fw + 6776832;        // 41943040 (feat / cat7 / h8pre)
  float* BUFB = fw + 48719872;       // 20971520 (hpre / h6 / h7)

  dim3 blk(256);
  auto nblk = [](size_t total) { return dim3((unsigned)((total + 255) / 256)); };
  auto gemm = [&](const float* A, const float* Wm, float* O_, int M, int Ci, int O) {
    k_gemm_wmma<<<dim3(O / 32, M / 256), blk, 0, stream>>>(A, Wm, O_, M, Ci, O);
  };
  auto stats = [&](const float* H, int M, int O) {
    k_bn_stats<<<dim3(O), blk, 0, stream>>>(H, M, O, MEAN, RSTD);
  };

  // inputs -> xt (and first output)
  k_transpose<<<nblk((size_t)B_ * N_ * C_), blk, 0, stream>>>(x, XT, out);

  // ---- Stage 1: knn on xt[...,6:9], edge(18)->w1(64)->w2(64)->maxK -> x1
  k_knn<3><<<dim3(B_ * (N_ / 256)), blk, 0, stream>>>(XT + 6, C_, IDX);
  k_edge_feat<9><<<nblk((size_t)BNK_ * 18), blk, 0, stream>>>(XT, IDX, BUFA);
  gemm(BUFA, W[0], BUFB, BNK_, 18, 64);
  stats(BUFB, BNK_, 64);
  k_bn_lrelu<<<nblk((size_t)BNK_ * 64), blk, 0, stream>>>(BUFB, 64, (size_t)BNK_ * 64,
                                                          MEAN, RSTD, G[0], Bt[0]);
  gemm(BUFB, W[1], BUFA, BNK_, 64, 64);
  stats(BUFA, BNK_, 64);
  k_bn_lrelu_maxk<<<nblk((size_t)BN_ * 64), blk, 0, stream>>>(BUFA, 64, MEAN, RSTD,
                                                              G[1], Bt[1], X1);

  // ---- Stage 2: knn on x1, edge(128)->w3(64)->w4(64)->maxK -> x2
  k_knn<64><<<dim3(B_ * (N_ / 256)), blk, 0, stream>>>(X1, 64, IDX);
  k_edge_feat<64><<<nblk((size_t)BNK_ * 128), blk, 0, stream>>>(X1, IDX, BUFA);
  gemm(BUFA, W[2], BUFB, BNK_, 128, 64);
  stats(BUFB, BNK_, 64);
  k_bn_lrelu<<<nblk((size_t)BNK_ * 64), blk, 0, stream>>>(BUFB, 64, (size_t)BNK_ * 64,
                                                          MEAN, RSTD, G[2], Bt[2]);
  gemm(BUFB, W[3], BUFA, BNK_, 64, 64);
  stats(BUFA, BNK_, 64);
  k_bn_lrelu_maxk<<<nblk((size_t)BN_ * 64), blk, 0, stream>>>(BUFA, 64, MEAN, RSTD,
                                                              G[3], Bt[3], X2);

  // ---- Stage 3: knn on x2, edge(128)->w5(64)->maxK -> x3
  k_knn<64><<<dim3(B_ * (N_ / 256)), blk, 0, stream>>>(X2, 64, IDX);
  k_edge_feat<64><<<nblk((size_t)BNK_ * 128), blk, 0, stream>>>(X2, IDX, BUFA);
  gemm(BUFA, W[4], BUFB, BNK_, 128, 64);
  stats(BUFB, BNK_, 64);
  k_bn_lrelu_maxk<<<nblk((size_t)BN_ * 64), blk, 0, stream>>>(BUFB, 64, MEAN, RSTD,
                                                              G[4], Bt[4], X3);

  // ---- Point MLPs: w6(1024) -> gmax -> w7(512) -> w8(256)
  k_cat3<<<nblk((size_t)BN_ * 192), blk, 0, stream>>>(X1, X2, X3, CAT6);
  gemm(CAT6, W[5], BUFB, BN_, 192, 1024);
  stats(BUFB, BN_, 1024);
  k_bn_lrelu<<<nblk((size_t)BN_ * 1024), blk, 0, stream>>>(BUFB, 1024, (size_t)BN_ * 1024,
                                                           MEAN, RSTD, G[5], Bt[5]);
  k_gmax<<<nblk((size_t)B_ * 1024), blk, 0, stream>>>(BUFB, GMAX);
  k_cat7<<<nblk((size_t)BN_ * 1216), blk, 0, stream>>>(GMAX, X1, X2, X3, BUFA);
  gemm(BUFA, W[6], BUFB, BN_, 1216, 512);
  stats(BUFB, BN_, 512);
  k_bn_lrelu<<<nblk((size_t)BN_ * 512), blk, 0, stream>>>(BUFB, 512, (size_t)BN_ * 512,
                                                          MEAN, RSTD, G[6], Bt[6]);
  gemm(BUFB, W[7], BUFA, BN_, 512, 256);
  stats(BUFA, BN_, 256);
  k_bn_lrelu_tr_out<<<nblk((size_t)BN_ * 256), blk, 0, stream>>>(BUFA, MEAN, RSTD,
                                                                 G[7], Bt[7], out + 147456);
}